// GAE_CLS_Link_NODE_Cosine_AttValue_32212254720634
// MI455X (gfx1250) — compile-verified
//
#include <hip/hip_runtime.h>
#include <math.h>

typedef __attribute__((ext_vector_type(2))) float v2f;
typedef __attribute__((ext_vector_type(8))) float v8f;

#define ACT_SLOPE 0.01f
#define GAT_SLOPE 0.2f
#define EPS_GN 1e-5f

// ---------------------------------------------------------------------------
// WMMA helpers: each wave computes 16 rows x 64 cols (4 accumulators), so one
// A fragment (16x4 f32) is reused by 4 chained V_WMMA_F32_16X16X4_F32 ops.
// ---------------------------------------------------------------------------
__device__ __forceinline__ void store_tile(
    float* __restrict__ out, const v8f& acc, const float* __restrict__ bvec,
    int rbase, int ncol, int N, bool act)
{
  float bias = bvec[ncol];
  #pragma unroll
  for (int v = 0; v < 8; ++v) {
    int row = rbase + v;
    if (row < N) {
      float val = acc[v] + bias;
      if (act) val = val > 0.f ? val : ACT_SLOPE * val;
      out[(size_t)row * 64 + ncol] = val;
    }
  }
}

// ---------------------------------------------------------------------------
// Encoder: x = leaky_relu(concat(x_feat, emb[node_ids]) @ w_in + b_in, 0.01)
// Block = 128 thr (4 waves) covering 64 rows; K = 128.
// ---------------------------------------------------------------------------
__global__ __launch_bounds__(128) void encode_kernel(
    const float* __restrict__ x_feat, const int* __restrict__ node_ids,
    const float* __restrict__ emb_table, const float* __restrict__ w_in,
    const float* __restrict__ b_in, float* __restrict__ xout, int N)
{
  __shared__ float xs[64][129];          // pad -> conflict-free column reads
  const int m0  = blockIdx.x * 64;
  const int tid = threadIdx.x;

  // prefetch next block's rows (global_prefetch_b8)
  {
    int nrow = m0 + 64 + (tid >> 1);
    if (nrow < N)
      __builtin_prefetch(x_feat + (size_t)nrow * 124 + (tid & 1) * 64, 0, 3);
  }

  for (int i = tid; i < 64 * 128; i += 128) {
    int r = i >> 7, k = i & 127;
    int row = m0 + r; if (row >= N) row = N - 1;
    float v = (k < 124) ? x_feat[(size_t)row * 124 + k]
                        : emb_table[node_ids[row] * 4 + (k - 124)];
    xs[r][k] = v;
  }
  __syncthreads();

  const int wave = tid >> 5;
  const int lane = tid & 31;
  const int col  = lane & 15;
  const int koff = (lane >> 4) << 1;            // lanes 16-31: K+2,K+3
  const int arow = (wave << 4) + (lane & 15);   // this wave's 16-row stripe

  v8f acc0 = {}, acc1 = {}, acc2 = {}, acc3 = {};
  for (int k0 = 0; k0 < 128; k0 += 4) {
    v2f a;
    a.x = xs[arow][k0 + koff];
    a.y = xs[arow][k0 + koff + 1];
    const float* w0 = w_in + (k0 + koff)     * 64 + col;
    const float* w1 = w_in + (k0 + koff + 1) * 64 + col;
    v2f b;
    b.x = w0[0];  b.y = w1[0];
    acc0 = __builtin_amdgcn_wmma_f32_16x16x4_f32(false, a, false, b, (short)0, acc0, false, false);
    b.x = w0[16]; b.y = w1[16];
    acc1 = __builtin_amdgcn_wmma_f32_16x16x4_f32(false, a, false, b, (short)0, acc1, false, false);
    b.x = w0[32]; b.y = w1[32];
    acc2 = __builtin_amdgcn_wmma_f32_16x16x4_f32(false, a, false, b, (short)0, acc2, false, false);
    b.x = w0[48]; b.y = w1[48];
    acc3 = __builtin_amdgcn_wmma_f32_16x16x4_f32(false, a, false, b, (short)0, acc3, false, false);
  }

  const int rbase = m0 + (wave << 4) + ((lane >> 4) << 3);
  store_tile(xout, acc0, b_in, rbase, 0  + col, N, true);
  store_tile(xout, acc1, b_in, rbase, 16 + col, N, true);
  store_tile(xout, acc2, b_in, rbase, 32 + col, N, true);
  store_tile(xout, acc3, b_in, rbase, 48 + col, N, true);
}

// ---------------------------------------------------------------------------
// Per-layer linear: g = x @ W + b   (K = 64, no activation)
// ---------------------------------------------------------------------------
__global__ __launch_bounds__(128) void gemm64_kernel(
    const float* __restrict__ xin, const float* __restrict__ W,
    const float* __restrict__ bvec, float* __restrict__ gout, int N)
{
  __shared__ float xs[64][65];
  const int m0  = blockIdx.x * 64;
  const int tid = threadIdx.x;

  for (int i = tid; i < 64 * 64; i += 128) {
    int r = i >> 6, k = i & 63;
    int row = m0 + r; if (row >= N) row = N - 1;
    xs[r][k] = xin[(size_t)row * 64 + k];
  }
  __syncthreads();

  const int wave = tid >> 5;
  const int lane = tid & 31;
  const int col  = lane & 15;
  const int koff = (lane >> 4) << 1;
  const int arow = (wave << 4) + (lane & 15);

  v8f acc0 = {}, acc1 = {}, acc2 = {}, acc3 = {};
  for (int k0 = 0; k0 < 64; k0 += 4) {
    v2f a;
    a.x = xs[arow][k0 + koff];
    a.y = xs[arow][k0 + koff + 1];
    const float* w0 = W + (k0 + koff)     * 64 + col;
    const float* w1 = W + (k0 + koff + 1) * 64 + col;
    v2f b;
    b.x = w0[0];  b.y = w1[0];
    acc0 = __builtin_amdgcn_wmma_f32_16x16x4_f32(false, a, false, b, (short)0, acc0, false, false);
    b.x = w0[16]; b.y = w1[16];
    acc1 = __builtin_amdgcn_wmma_f32_16x16x4_f32(false, a, false, b, (short)0, acc1, false, false);
    b.x = w0[32]; b.y = w1[32];
    acc2 = __builtin_amdgcn_wmma_f32_16x16x4_f32(false, a, false, b, (short)0, acc2, false, false);
    b.x = w0[48]; b.y = w1[48];
    acc3 = __builtin_amdgcn_wmma_f32_16x16x4_f32(false, a, false, b, (short)0, acc3, false, false);
  }

  const int rbase = m0 + (wave << 4) + ((lane >> 4) << 3);
  store_tile(gout, acc0, bvec, rbase, 0  + col, N, false);
  store_tile(gout, acc1, bvec, rbase, 16 + col, N, false);
  store_tile(gout, acc2, bvec, rbase, 32 + col, N, false);
  store_tile(gout, acc3, bvec, rbase, 48 + col, N, false);
}

// ---------------------------------------------------------------------------
// Edge helpers
// ---------------------------------------------------------------------------
__device__ __forceinline__ void edge_pair(const int* __restrict__ ei, int E,
                                          int i, int& s, int& d) {
  if (i < E) { s = ei[i]; d = ei[E + i]; }
  else       { s = i - E; d = i - E; }           // appended self loops
}

__device__ __forceinline__ void atomicMaxFloat(float* addr, float value) {
  if (value >= 0.f)
    atomicMax((int*)addr, __float_as_int(value));
  else
    atomicMin((unsigned int*)addr, __float_as_uint(value));
}

// m = -inf, s = 0
__global__ __launch_bounds__(256) void init_ms_kernel(
    float* __restrict__ m, float* __restrict__ s, int N)
{
  int i = blockIdx.x * blockDim.x + threadIdx.x;
  if (i < N) { m[i] = -INFINITY; s[i] = 0.f; }
}

// e[i] = dot(leaky(g[src]+g[dst], 0.2), att); segment max into m[dst]
__global__ __launch_bounds__(256) void edge_score_kernel(
    const float* __restrict__ g, const int* __restrict__ ei, int E, int nE,
    const float* __restrict__ att, float* __restrict__ e, float* __restrict__ m)
{
  int widx = (blockIdx.x * blockDim.x + threadIdx.x) >> 5;
  int lane = threadIdx.x & 31;
  if (widx >= nE) return;
  int s, d; edge_pair(ei, E, widx, s, d);

  float2 a  = ((const float2*)(g + (size_t)s * 64))[lane];
  float2 b  = ((const float2*)(g + (size_t)d * 64))[lane];
  float2 at = ((const float2*)att)[lane];
  float v0 = a.x + b.x; v0 = v0 > 0.f ? v0 : GAT_SLOPE * v0;
  float v1 = a.y + b.y; v1 = v1 > 0.f ? v1 : GAT_SLOPE * v1;
  float p  = v0 * at.x + v1 * at.y;
  #pragma unroll
  for (int off = 16; off; off >>= 1) p += __shfl_xor(p, off, 32);
  if (lane == 0) { e[widx] = p; atomicMaxFloat(&m[d], p); }
}

// e[i] = exp(e[i] - m[dst]); segment sum into s[dst]
__global__ __launch_bounds__(256) void exp_sum_kernel(
    float* __restrict__ e, const float* __restrict__ m,
    float* __restrict__ ssum, const int* __restrict__ ei, int E, int nE)
{
  int i = blockIdx.x * blockDim.x + threadIdx.x;
  if (i >= nE) return;
  int s, d; edge_pair(ei, E, i, s, d);
  float v = __expf(e[i] - m[d]);
  e[i] = v;
  atomicAdd(&ssum[d], v);
}

// alpha = ee/s[dst] -> out; xacc[dst] += alpha * g[src]
__global__ __launch_bounds__(256) void aggregate_kernel(
    const float* __restrict__ g, const int* __restrict__ ei, int E, int nE,
    const float* __restrict__ ee, const float* __restrict__ ssum,
    float* __restrict__ alpha_out, float* __restrict__ xacc)
{
  int widx = (blockIdx.x * blockDim.x + threadIdx.x) >> 5;
  int lane = threadIdx.x & 31;
  if (widx >= nE) return;
  int s, d; edge_pair(ei, E, widx, s, d);
  float alpha = ee[widx] / ssum[d];
  if (lane == 0) alpha_out[widx] = alpha;
  float2 gv = ((const float2*)(g + (size_t)s * 64))[lane];
  float* dstp = xacc + (size_t)d * 64 + 2 * lane;
  atomicAdd(dstp,     alpha * gv.x);
  atomicAdd(dstp + 1, alpha * gv.y);
}

// ---------------------------------------------------------------------------
// GraphNorm: per-column sum / sum-of-squares of the aggregated x
// ---------------------------------------------------------------------------
__global__ __launch_bounds__(256) void stats_kernel(
    const float* __restrict__ x, float* __restrict__ colsum,
    float* __restrict__ colsq, int total)
{
  __shared__ float ls[64], lq[64];
  if (threadIdx.x < 64) { ls[threadIdx.x] = 0.f; lq[threadIdx.x] = 0.f; }
  __syncthreads();
  int stride = gridDim.x * blockDim.x;              // multiple of 64
  float s = 0.f, q = 0.f;
  for (int i = blockIdx.x * blockDim.x + threadIdx.x; i < total; i += stride) {
    float v = x[i]; s += v; q += v * v;
  }
  int c = threadIdx.x & 63;
  atomicAdd(&ls[c], s); atomicAdd(&lq[c], q);
  __syncthreads();
  if (threadIdx.x < 64) {
    atomicAdd(&colsum[threadIdx.x], ls[threadIdx.x]);
    atomicAdd(&colsq[threadIdx.x],  lq[threadIdx.x]);
  }
}

// Fold conv_b + GraphNorm into per-column affine: out = xagg*A + B
__global__ __launch_bounds__(64) void coeffs_kernel(
    const float* __restrict__ colsum, const float* __restrict__ colsq,
    const float* __restrict__ conv_b, const float* __restrict__ gn_w,
    const float* __restrict__ gn_b, const float* __restrict__ gn_ms,
    float* __restrict__ A, float* __restrict__ B, int N)
{
  int c = threadIdx.x;
  float ninv  = 1.f / (float)N;
  float Ex    = colsum[c] * ninv;                  // E[xagg]
  float Exx   = colsq[c]  * ninv;                  // E[xagg^2]
  float cb    = conv_b[c];
  float meanY = Ex + cb;                           // E[y], y = xagg + cb
  float Eyy   = Exx + 2.f * cb * Ex + cb * cb;     // E[y^2]
  float mm    = meanY * gn_ms[c];
  float var   = Eyy - 2.f * mm * meanY + mm * mm;  // E[(y-mm)^2]
  float inv   = rsqrtf(var + EPS_GN);
  float Ac    = inv * gn_w[c];
  A[c] = Ac;
  B[c] = (cb - mm) * Ac + gn_b[c];
}

// x = leaky(x*A + B, 0.01);  h += 0.5*x
__global__ __launch_bounds__(256) void norm_act_kernel(
    float* __restrict__ x, float* __restrict__ h,
    const float* __restrict__ A, const float* __restrict__ B, int total)
{
  int i = blockIdx.x * blockDim.x + threadIdx.x;
  if (i >= total) return;
  int c = i & 63;
  float v = x[i] * A[c] + B[c];
  v = v > 0.f ? v : ACT_SLOPE * v;
  x[i] = v;
  h[i] += 0.5f * v;
}

// ---------------------------------------------------------------------------
extern "C" void kernel_launch(void* const* d_in, const int* in_sizes, int n_in,
                              void* d_out, int out_size, void* d_ws, size_t ws_size,
                              hipStream_t stream)
{
  const float* x_feat     = (const float*)d_in[0];
  const int*   node_ids   = (const int*)  d_in[1];
  const int*   edge_index = (const int*)  d_in[2];   // [2,E]
  const float* emb_table  = (const float*)d_in[3];
  const float* w_in       = (const float*)d_in[4];
  const float* b_in       = (const float*)d_in[5];
  const float* lin_w      = (const float*)d_in[6];   // [L,64,64]
  const float* lin_b      = (const float*)d_in[7];   // [L,64]
  const float* att        = (const float*)d_in[8];   // [L,64]
  const float* conv_b     = (const float*)d_in[9];
  const float* gn_w       = (const float*)d_in[10];
  const float* gn_b       = (const float*)d_in[11];
  const float* gn_ms      = (const float*)d_in[12];

  const int N  = in_sizes[1];
  const int E  = in_sizes[2] / 2;
  const int nE = E + N;
  const int L  = 4;

  // Output layout: x [N,64] | h [N,64] | xs [L, nE]
  float* out_x  = (float*)d_out;
  float* out_h  = out_x + (size_t)N * 64;
  float* out_xs = out_h + (size_t)N * 64;

  // Workspace layout (floats)
  float* g      = (float*)d_ws;              // N*64
  float* ebuf   = g      + (size_t)N * 64;   // nE  (e, then ee in-place)
  float* mbuf   = ebuf   + nE;               // N
  float* sbuf   = mbuf   + N;                // N
  float* colsum = sbuf   + N;                // 64
  float* colsq  = colsum + 64;               // 64
  float* Acoef  = colsq  + 64;               // 64
  float* Bcoef  = Acoef  + 64;               // 64

  const int mblocks = (N + 63) / 64;         // 64 rows per block
  const int eblocks = (nE + 7) / 8;          // 8 waves (edges) per 256-thr block

  hipMemsetAsync(out_h, 0, (size_t)N * 64 * sizeof(float), stream);

  encode_kernel<<<mblocks, 128, 0, stream>>>(
      x_feat, node_ids, emb_table, w_in, b_in, out_x, N);

  for (int l = 0; l < L; ++l) {
    gemm64_kernel<<<mblocks, 128, 0, stream>>>(
        out_x, lin_w + (size_t)l * 64 * 64, lin_b + l * 64, g, N);

    init_ms_kernel<<<(N + 255) / 256, 256, 0, stream>>>(mbuf, sbuf, N);
    hipMemsetAsync(colsum, 0, 128 * sizeof(float), stream);
    hipMemsetAsync(out_x, 0, (size_t)N * 64 * sizeof(float), stream);

    edge_score_kernel<<<eblocks, 256, 0, stream>>>(
        g, edge_index, E, nE, att + l * 64, ebuf, mbuf);

    exp_sum_kernel<<<(nE + 255) / 256, 256, 0, stream>>>(
        ebuf, mbuf, sbuf, edge_index, E, nE);

    aggregate_kernel<<<eblocks, 256, 0, stream>>>(
        g, edge_index, E, nE, ebuf, sbuf, out_xs + (size_t)l * nE, out_x);

    stats_kernel<<<512, 256, 0, stream>>>(out_x, colsum, colsq, N * 64);

    coeffs_kernel<<<1, 64, 0, stream>>>(
        colsum, colsq, conv_b + l * 64, gn_w + l * 64, gn_b + l * 64,
        gn_ms + l * 64, Acoef, Bcoef, N);

    norm_act_kernel<<<(N * 64 + 255) / 256, 256, 0, stream>>>(
        out_x, out_h, Acoef, Bcoef, N * 64);
  }
}